// CHTransform_43439299231904
// MI455X (gfx1250) — compile-verified
//
#include <hip/hip_runtime.h>
#include <hip/hip_bf16.h>

typedef __attribute__((ext_vector_type(2))) float v2f;
typedef __attribute__((ext_vector_type(8))) float v8f;

#define PI_F 3.14159265358979323846f

// Bessel zeros table (|n| = 0..3, k = 1..4)
__device__ __constant__ float c_bz[4][4] = {
    {2.4048f, 5.5201f, 8.6537f, 11.7915f},
    {3.8317f, 7.0156f, 10.1735f, 13.3237f},
    {5.1356f, 8.4172f, 11.6198f, 14.796f},
    {6.3802f, 9.761f, 13.0152f, 16.2235f}};

__device__ __forceinline__ float bessel_j(int n, float x) {
    float xs = fmaxf(x, 1e-12f);
    if (n == 0) {
        float sm = 1.f - x * x * 0.25f + x * x * x * x * (1.f / 64.f);
        float md = cosf(x - PI_F * 0.25f) * rsqrtf(xs);
        float lg = sqrtf(2.f / (PI_F * xs)) * cosf(x - PI_F * 0.25f);
        float ax = fabsf(x);
        return (ax < 1.f) ? sm : ((ax < 5.f) ? md : lg);
    } else if (n == 1) {
        float sm = x * 0.5f - x * x * x * (1.f / 16.f);
        float md = sinf(x - PI_F * 0.25f) * rsqrtf(xs);
        float lg = sqrtf(2.f / (PI_F * xs)) * cosf(x - 3.f * PI_F * 0.25f);
        float ax = fabsf(x);
        return (ax < 1.f) ? sm : ((ax < 5.f) ? md : lg);
    } else {
        float logfact = (n == 2) ? 0.69314718056f : 1.79175946923f; // log(2!), log(3!)
        float sm = expf((float)n * logf(xs * 0.5f) - logfact);
        float lg = sqrtf(2.f / (PI_F * xs)) * cosf(x - (float)(2 * n + 1) * PI_F * 0.25f);
        return (fabsf(x) < 0.1f * (float)n) ? sm : lg;
    }
}

// ---------------------------------------------------------------------------
// Kernel 0: basis tables.
//   axBt [32][96] : TRANSPOSED axial B-matrix; col c = 2l(+1) -> cos/sin(pi*lv*z)/sqrt(2)
//   ang  [7][96][2], Wr [7][4][96]
// ---------------------------------------------------------------------------
__global__ __launch_bounds__(256) void ch_setup(float* __restrict__ axBt,
                                                float* __restrict__ ang,
                                                float* __restrict__ Wr) {
    int tid = threadIdx.x;
    for (int i = tid; i < 32 * 96; i += 256) {
        int c = i / 96, z = i % 96;
        float val = 0.f;
        if (c < 22) {
            int l = c >> 1;
            float lv = (float)(l - 5);
            float zz = -1.f + 2.f * (float)z / 95.f;
            float ph = PI_F * lv * zz;
            val = ((c & 1) ? sinf(ph) : cosf(ph)) * 0.70710678118f; // 1/sqrt(2)
        }
        axBt[i] = val;
    }
    for (int i = tid; i < 7 * 96; i += 256) {
        int n = i / 96, t = i % 96;
        float nv = (float)(n - 3);
        float th = 2.f * PI_F * (float)t / 95.f;
        ang[i * 2 + 0] = cosf(nv * th) * 0.39894228040f; // 1/sqrt(2pi)
        ang[i * 2 + 1] = sinf(nv * th) * 0.39894228040f;
    }
    if (tid < 28) {
        int n = tid / 4, k = tid % 4;
        int an = abs(n - 3);
        float rnk = c_bz[an][k];
        float dr = 1.f / 95.f;
        float ss = 0.f;
        for (int j = 0; j < 96; j++) {
            float rj = (float)j / 95.f;
            float J = bessel_j(an, rnk * rj);
            float term = J * rj * dr;
            ss += term * term;
        }
        ss *= 96.f * 96.f; // T*Z factor
        float norm = (ss > 1e-6f) ? rsqrtf(ss) : 0.f;
        for (int j = 0; j < 96; j++) {
            float rj = (float)j / 95.f;
            float J = bessel_j(an, rnk * rj);
            Wr[(n * 4 + k) * 96 + j] = J * norm * rj * dr;
        }
    }
}

// ---------------------------------------------------------------------------
// Kernel 1: fused stage 1 (Z-contraction, WMMA f32) + stage 2 (T-contraction).
// One workgroup per (b,c,r); 6 waves; wave w owns M rows [16w,16w+16), both
// N-tiles. vol tile streamed with GLOBAL_LOAD_ASYNC_TO_LDS_B128 (ASYNCcnt).
// ---------------------------------------------------------------------------
#define VPITCH 100 // conflict-free pitch for fragment gathers (gcd-free mod 64)
#define BPITCH 100
#define APITCH 33

__global__ __launch_bounds__(192) void ch_stage12(const float* __restrict__ vol,
                                                  const float* __restrict__ axBt,
                                                  const float* __restrict__ ang,
                                                  float* __restrict__ Bc) {
    __shared__ float sV[96 * VPITCH];  // 38400 B  vol tile [t][z]; reused as sA after WMMA
    __shared__ float sBt[32 * BPITCH]; // 12800 B  transposed axial [col][z]
    __shared__ float sAng[7 * 96 * 2]; //  5376 B  (total 56576 B)

    const int bcr = blockIdx.x; // (b*8+c)*96 + r
    const int tid = threadIdx.x;
    const float* volBase = vol + (size_t)bcr * (96 * 96);

    // --- async-copy the 36 KB vol tile straight into LDS (no VGPR staging) ---
    for (int i = tid; i < 96 * 96 / 4; i += 192) {
        int t = (i * 4) / 96, z = (i * 4) % 96;
        unsigned lds_off = (unsigned)(size_t)(&sV[t * VPITCH + z]); // LDS byte offset
        const float* g = volBase + i * 4;
        asm volatile("global_load_async_to_lds_b128 %0, %1, off"
                     :
                     : "v"(lds_off), "v"(g)
                     : "memory");
    }
    // --- small tables via regular path ---
    for (int i = tid; i < 32 * 96 / 4; i += 192) {
        int c = (i * 4) / 96, z = (i * 4) % 96;
        *(float4*)&sBt[c * BPITCH + z] = ((const float4*)axBt)[i];
    }
    for (int i = tid; i < 7 * 96 * 2; i += 192)
        sAng[i] = ang[i];
    asm volatile("s_wait_asynccnt 0x0" ::: "memory");
    __syncthreads();

    // ---- stage 1: A[t][l2] = sum_z sV[t][z] * axial[z][l2]  (M=96,K=96,N=22->32)
    const int wave = tid >> 5;
    const int lane = tid & 31;
    const int m0 = wave * 16;
    const int lrow = lane & 15;
    const int khalf = (lane >> 4) << 1; // 0 or 2

    v8f c0 = {};
    v8f c1 = {};
#pragma unroll 4
    for (int kk = 0; kk < 96; kk += 4) {
        v2f a = *(const v2f*)&sV[(m0 + lrow) * VPITCH + kk + khalf];
        v2f b0 = *(const v2f*)&sBt[lrow * BPITCH + kk + khalf];
        v2f b1 = *(const v2f*)&sBt[(16 + lrow) * BPITCH + kk + khalf];
        c0 = __builtin_amdgcn_wmma_f32_16x16x4_f32(false, a, false, b0,
                                                   (short)0, c0, false, false);
        c1 = __builtin_amdgcn_wmma_f32_16x16x4_f32(false, a, false, b1,
                                                   (short)0, c1, false, false);
    }

    // vol tile dead; reuse its LDS as sA[96][APITCH] (branch-free D store)
    __syncthreads();
    float* sA = sV;
    {
        const int drow = m0 + ((lane >> 4) << 3); // D: VGPR v -> M = v (+8 for hi half)
        const int dcol = lrow;
#pragma unroll
        for (int v = 0; v < 8; v++) {
            sA[(drow + v) * APITCH + dcol] = c0[v];
            sA[(drow + v) * APITCH + 16 + dcol] = c1[v];
        }
    }
    __syncthreads();

    // ---- stage 2: Bc[n][l] = sum_t ang[n][t] * A[t][l]  (complex x complex)
    if (tid < 77) {
        int n = tid / 11, l = tid % 11;
        float brs = 0.f, bis = 0.f;
        for (int t = 0; t < 96; t++) {
            float ar = sA[t * APITCH + 2 * l];
            float ai = sA[t * APITCH + 2 * l + 1];
            float cR = sAng[(n * 96 + t) * 2 + 0];
            float cI = sAng[(n * 96 + t) * 2 + 1];
            brs += ar * cR - ai * cI;
            bis += ar * cI + ai * cR;
        }
        float* o = Bc + (size_t)bcr * 160 + (size_t)(n * 11 + l) * 2;
        o[0] = brs;
        o[1] = bis;
    }
}

// ---------------------------------------------------------------------------
// Kernel 2: stage 3 — ch[b,c,n,k,l] = dtheta*dz * sum_r Bc[bc,r,n,l]*Wr[n,k,r]
// ---------------------------------------------------------------------------
__global__ __launch_bounds__(256) void ch_stage3(const float* __restrict__ Bc,
                                                 const float* __restrict__ Wr,
                                                 float* __restrict__ out) {
    const int NOUT = 8 * 8 * 7 * 4 * 11; // complex elements
    int idx = blockIdx.x * 256 + threadIdx.x;
    if (idx >= NOUT) return;
    int l = idx % 11;
    int t1 = idx / 11;
    int k = t1 % 4;
    int t2 = t1 / 4;
    int n = t2 % 7;
    int bc = t2 / 7;
    const float* bb = Bc + (size_t)bc * 96 * 160 + (size_t)(n * 11 + l) * 2;
    const float* w = Wr + (n * 4 + k) * 96;
    float sr = 0.f, si = 0.f;
    for (int r = 0; r < 96; r++) {
        float wv = w[r];
        sr += wv * bb[(size_t)r * 160];
        si += wv * bb[(size_t)r * 160 + 1];
    }
    const float scale = (2.f * PI_F / 96.f) * (2.f / 95.f); // dtheta * dz
    out[idx * 2 + 0] = sr * scale;
    out[idx * 2 + 1] = si * scale;
}

extern "C" void kernel_launch(void* const* d_in, const int* in_sizes, int n_in,
                              void* d_out, int out_size, void* d_ws, size_t ws_size,
                              hipStream_t stream) {
    (void)in_sizes; (void)n_in; (void)out_size; (void)ws_size;
    const float* vol = (const float*)d_in[0];
    float* out = (float*)d_out; // complex64 -> interleaved re/im floats

    char* ws = (char*)d_ws;
    float* axBt = (float*)(ws);                 // 32*96*4   = 12288 B
    float* ang  = (float*)(ws + 12288);         // 7*96*2*4  =  5376 B
    float* Wr   = (float*)(ws + 12288 + 5376);  // 7*4*96*4  = 10752 B
    float* Bc   = (float*)(ws + 28672);         // 6144*160*4 = 3,932,160 B

    ch_setup<<<1, 256, 0, stream>>>(axBt, ang, Wr);
    ch_stage12<<<8 * 8 * 96, 192, 0, stream>>>(vol, axBt, ang, Bc);
    const int NOUT = 8 * 8 * 7 * 4 * 11;
    ch_stage3<<<(NOUT + 255) / 256, 256, 0, stream>>>(Bc, Wr, out);
}